// AllocortexSystem_40192303956593
// MI455X (gfx1250) — compile-verified
//
#include <hip/hip_runtime.h>
#include <cstdint>
#include <math.h>

#define DIMD 1024
#define NROWS 262144
#define TILE_ROWS 16
#define ROW_F 1028                      // padded LDS row stride in floats (bank-conflict-free A-frags)
#define NTILES (NROWS / TILE_ROWS)      // 16384
#define GPASS 512
#define TILES_PER_WG (NTILES / GPASS)   // 32
#define DG_THR 0.1f
#define EPS 1e-8f

typedef float v2f __attribute__((ext_vector_type(2)));
typedef float v8f __attribute__((ext_vector_type(8)));
typedef unsigned int u32x4 __attribute__((ext_vector_type(4)));
typedef int i32x4 __attribute__((ext_vector_type(4)));
typedef int i32x8 __attribute__((ext_vector_type(8)));

// ---- TDM: one descriptor loads 16 rows (4KB each) into LDS with padded row
// stride via iterate mode (16 iterations, lds += 1028 floats, global += 1024). ----
__device__ __forceinline__ void tdm_load_tile(const float* gsrc, uint32_t lds_byte_off) {
  uint64_t ga = (uint64_t)(uintptr_t)gsrc;
  u32x4 g0;
  g0[0] = 1u;                                            // count=1, user mode
  g0[1] = lds_byte_off;                                  // lds_addr
  g0[2] = (uint32_t)ga;                                  // global_addr[31:0]
  g0[3] = (uint32_t)((ga >> 32) & 0x01FFFFFFull)         // global_addr[56:32]
          | 0x80000000u;                                 // type=2 ("image")
  i32x8 g1;
  g1[0] = (2 << 16) | (1 << 19);                         // data_size=4B, iterate_enable=1
  g1[1] = (int)(1024u << 16);                            // tensor_dim0 = 1024 (lo16)
  g1[2] = 0;                                             // dim0 hi / dim1 lo (262144 -> lo16=0)
  g1[3] = (int)(4u | (1024u << 16));                     // tensor_dim1 hi=4, tile_dim0=1024
  g1[4] = 1;                                             // tile_dim1=1, tile_dim2=0
  g1[5] = 1024;                                          // tensor_dim0_stride lo
  g1[6] = 0;
  g1[7] = 0;
  i32x4 g2;
  g2[0] = 0;
  g2[1] = ROW_F;                                         // lds_addr_increment (elements)
  g2[2] = 1024;                                          // global_addr_increment (elements)
  g2[3] = (TILE_ROWS - 1) << 16;                         // iterate_count = 15 -> 16 rows
  i32x4 g3; g3[0] = 0; g3[1] = 0; g3[2] = 0; g3[3] = 0;
  i32x8 g4; g4[0] = 0; g4[1] = 0; g4[2] = 0; g4[3] = 0;
  g4[4] = 0; g4[5] = 0; g4[6] = 0; g4[7] = 0;            // unused VADDR4 slot
  __builtin_amdgcn_tensor_load_to_lds(g0, g1, g2, g3, g4, 0);
}

// ---- prep: sparse_prior -> cur ----
__global__ void k_prep(const float* __restrict__ iso, float* __restrict__ cur) {
  int i = threadIdx.x;
  float4 v = ((const float4*)iso)[i];
  v.x = (v.x > DG_THR) ? v.x : 0.f;
  v.y = (v.y > DG_THR) ? v.y : 0.f;
  v.z = (v.z > DG_THR) ? v.z : 0.f;
  v.w = (v.w > DG_THR) ? v.w : 0.f;
  ((float4*)cur)[i] = v;
}

// ---- q-norm ----
__global__ void k_qnorm(const float* __restrict__ cur, float* __restrict__ qn) {
  __shared__ float red[256];
  int i = threadIdx.x;
  float4 v = ((const float4*)cur)[i];
  red[i] = v.x * v.x + v.y * v.y + v.z * v.z + v.w * v.w;
  __syncthreads();
  for (int off = 128; off > 0; off >>= 1) {
    if (i < off) red[i] += red[i + off];
    __syncthreads();
  }
  if (i == 0) qn[0] = fmaxf(sqrtf(red[0]), EPS);
}

// ---- fused pass: per tile -> WMMA dots + WMMA row-norms + online softmax ----
__global__ void __launch_bounds__(256)
k_pass(const float* __restrict__ ca3, const float* __restrict__ cur,
       const float* __restrict__ qn, float* __restrict__ pm,
       float* __restrict__ ps, float* __restrict__ pacc) {
  extern __shared__ float smem[];
  float* buf0 = smem;
  float* buf1 = smem + TILE_ROWS * ROW_F;
  float* qlds = smem + 2 * TILE_ROWS * ROW_F;
  float* simpart = qlds + DIMD;   // [8][16]
  float* sqpart  = simpart + 128; // [8][16]

  const int tid  = threadIdx.x;
  const int w    = tid >> 5;
  const int lane = tid & 31;
  const int wg   = blockIdx.x;
  const int mycol = w * 128 + lane * 4;

  // stage q into LDS (feeds WMMA B-fragments)
  ((float4*)qlds)[tid] = ((const float4*)cur)[tid];
  const float qnv = qn[0];
  __syncthreads();

  const int rowSel = lane & 15;
  const int koff = (lane < 16) ? 0 : 2;

  // hoist tile-invariant q B-fragments into registers
  v2f bq[32];
#pragma unroll
  for (int kk = 0; kk < 32; ++kk)
    bq[kk] = *(const v2f*)(qlds + w * 128 + kk * 4 + koff);
  const v2f bones = {1.f, 1.f};

  float m = -3.0e38f, s = 0.f;
  float4 acc = make_float4(0.f, 0.f, 0.f, 0.f);

  if (w == 0)
    tdm_load_tile(ca3 + (size_t)wg * TILE_ROWS * DIMD, (uint32_t)(uintptr_t)buf0);

  for (int t = 0; t < TILES_PER_WG; ++t) {
    float* T = (t & 1) ? buf1 : buf0;
    if (w == 0) {
      if (t + 1 < TILES_PER_WG) {
        const int nt = wg + (t + 1) * GPASS;
        float* nb = (t & 1) ? buf0 : buf1;
        tdm_load_tile(ca3 + (size_t)nt * TILE_ROWS * DIMD, (uint32_t)(uintptr_t)nb);
        __builtin_amdgcn_s_wait_tensorcnt(1);
      } else {
        __builtin_amdgcn_s_wait_tensorcnt(0);
      }
    }
    __syncthreads();

    // --- Two independent WMMA chains over K-slice [w*128, w*128+128):
    //     c  = dot(row, q)   : A x q-broadcast-B
    //     c2 = sumsq(row)    : (A .* A) x ones-B   (replaces shuffle reductions)
    // A 16x4 f32 frag: lanes 0-15 -> M=lane, K=k0,k0+1 ; lanes 16-31 -> M=lane-16, K=k0+2,k0+3
    v8f c  = {0.f, 0.f, 0.f, 0.f, 0.f, 0.f, 0.f, 0.f};
    v8f c2 = {0.f, 0.f, 0.f, 0.f, 0.f, 0.f, 0.f, 0.f};
    const float* arow = T + rowSel * ROW_F + koff;
#pragma unroll
    for (int kk = 0; kk < 32; ++kk) {
      const int k0 = w * 128 + kk * 4;
      v2f a = *(const v2f*)(arow + k0);
      c = __builtin_amdgcn_wmma_f32_16x16x4_f32(false, a, false, bq[kk], (short)0, c,
                                                false, false);
      v2f a2 = a * a;
      c2 = __builtin_amdgcn_wmma_f32_16x16x4_f32(false, a2, false, bones, (short)0, c2,
                                                 false, false);
    }

    // C/D layout: VGPR i holds M=i (lanes 0-15) and M=8+i (lanes 16-31)
    if (lane == 0) {
#pragma unroll
      for (int i = 0; i < 8; ++i) {
        simpart[w * 16 + i] = c[i];
        sqpart[w * 16 + i]  = c2[i];
      }
    } else if (lane == 16) {
#pragma unroll
      for (int i = 0; i < 8; ++i) {
        simpart[w * 16 + 8 + i] = c[i];
        sqpart[w * 16 + 8 + i]  = c2[i];
      }
    }
    __syncthreads();

    // --- block-uniform online softmax; accumulator column-partitioned per wave ---
#pragma unroll 4
    for (int r = 0; r < TILE_ROWS; ++r) {
      float dot = 0.f, ss = 0.f;
#pragma unroll
      for (int ww = 0; ww < 8; ++ww) {
        dot += simpart[ww * 16 + r];
        ss  += sqpart[ww * 16 + r];
      }
      float nrm = fmaxf(sqrtf(ss), EPS);
      float sim = dot / (nrm * qnv);
      float wgt;
      if (sim > m) {                       // block-uniform branch
        float scale = __expf(m - sim);
        m = sim;
        s *= scale;
        acc.x *= scale; acc.y *= scale; acc.z *= scale; acc.w *= scale;
        wgt = 1.0f;
      } else {
        wgt = __expf(sim - m);
      }
      s += wgt;
      float4 x = *(const float4*)(T + r * ROW_F + mycol);
      acc.x += wgt * x.x; acc.y += wgt * x.y;
      acc.z += wgt * x.z; acc.w += wgt * x.w;
    }
    __syncthreads();
  }

  *(float4*)(pacc + (size_t)wg * DIMD + mycol) = acc;
  if (tid == 0) { pm[wg] = m; ps[wg] = s; }
}

// ---- global (max, sum) over per-WG partials ----
__global__ void k_ms(const float* __restrict__ pm, const float* __restrict__ ps,
                     float* __restrict__ ms2) {
  __shared__ float red[256];
  int i = threadIdx.x;
  float mloc = fmaxf(pm[i], pm[i + 256]);
  red[i] = mloc;
  __syncthreads();
  for (int off = 128; off > 0; off >>= 1) {
    if (i < off) red[i] = fmaxf(red[i], red[i + off]);
    __syncthreads();
  }
  float mstar = red[0];
  __syncthreads();
  float sl = ps[i] * __expf(pm[i] - mstar) + ps[i + 256] * __expf(pm[i + 256] - mstar);
  red[i] = sl;
  __syncthreads();
  for (int off = 128; off > 0; off >>= 1) {
    if (i < off) red[i] += red[i + off];
    __syncthreads();
  }
  if (i == 0) { ms2[0] = mstar; ms2[1] = red[0]; }
}

// ---- retrieved + state update ----
__global__ void k_update(const float* __restrict__ pacc, const float* __restrict__ pm,
                         const float* __restrict__ ms2, float* __restrict__ cur) {
  int d = blockIdx.x * 256 + threadIdx.x;
  float mstar = ms2[0];
  float S = ms2[1];
  float r = 0.f;
  for (int g = 0; g < GPASS; ++g)
    r += pacc[(size_t)g * DIMD + d] * __expf(pm[g] - mstar);
  float ret = r / S;
  cur[d] = 0.8f * ret + 0.2f * cur[d];
}

// ---- finalize: copy current + MSE ----
__global__ void k_final(const float* __restrict__ iso, const float* __restrict__ cur,
                        float* __restrict__ out) {
  __shared__ float red[256];
  int i = threadIdx.x;
  float4 c4 = ((const float4*)cur)[i];
  ((float4*)out)[i] = c4;
  float4 a = ((const float4*)iso)[i];
  float dx = a.x - c4.x, dy = a.y - c4.y, dz = a.z - c4.z, dw = a.w - c4.w;
  red[i] = dx * dx + dy * dy + dz * dz + dw * dw;
  __syncthreads();
  for (int off = 128; off > 0; off >>= 1) {
    if (i < off) red[i] += red[i + off];
    __syncthreads();
  }
  if (i == 0) out[DIMD] = red[0] / (float)DIMD;
}

extern "C" void kernel_launch(void* const* d_in, const int* in_sizes, int n_in,
                              void* d_out, int out_size, void* d_ws, size_t ws_size,
                              hipStream_t stream) {
  const float* iso = (const float*)d_in[0];
  const float* ca3 = (const float*)d_in[1];
  float* out = (float*)d_out;

  float* W    = (float*)d_ws;
  float* cur  = W;               // 1024
  float* qn   = W + 1024;        // 1
  float* ms2  = W + 1028;        // 2
  float* pm   = W + 1088;        // GPASS
  float* ps   = pm + GPASS;      // GPASS
  float* pacc = ps + GPASS;      // GPASS * 1024 (16B aligned: float off 2112)

  const size_t shbytes = (size_t)(2 * TILE_ROWS * ROW_F + DIMD + 256) * sizeof(float);

  k_prep<<<1, 256, 0, stream>>>(iso, cur);
  for (int step = 0; step < 5; ++step) {
    k_qnorm<<<1, 256, 0, stream>>>(cur, qn);
    k_pass<<<GPASS, 256, shbytes, stream>>>(ca3, cur, qn, pm, ps, pacc);
    k_ms<<<1, 256, 0, stream>>>(pm, ps, ms2);
    k_update<<<4, 256, 0, stream>>>(pacc, pm, ms2, cur);
  }
  k_final<<<1, 256, 0, stream>>>(iso, cur, out);
}